// LazySkipConnectionConvolutionalLayer_82411832475937
// MI455X (gfx1250) — compile-verified
//
#include <hip/hip_runtime.h>

#define D 32

typedef __attribute__((ext_vector_type(2))) float v2f;
typedef __attribute__((ext_vector_type(8))) float v8f;

// ---------------------------------------------------------------------------
// Kernel 1: zero the F1 accumulator in workspace (float4 stores)
// ---------------------------------------------------------------------------
__global__ __launch_bounds__(256) void zero_f32(float4* __restrict__ p, int n4) {
  int i = blockIdx.x * blockDim.x + threadIdx.x;
  if (i < n4) {
    float4 z = {0.f, 0.f, 0.f, 0.f};
    p[i] = z;
  }
}

// ---------------------------------------------------------------------------
// Kernel 2: edge scatter-add.  One wave per edge iteration:
//   lane l handles feature l of the 32-wide row.
//   src/dst indices are wave-uniform -> scalar (s_load) loads.
//   gather:  one coalesced 128B line per edge (L2-resident x)
//   scatter: one 128B-line worth of global_atomic_add_f32 per edge
// ---------------------------------------------------------------------------
__global__ __launch_bounds__(256) void edge_scatter(
    const float* __restrict__ x,
    const long long* __restrict__ ei,   // [2, n_edges] int64, row0=src row1=dst
    float* __restrict__ F1,
    int n_edges) {
  const int lane = threadIdx.x & 31;
  const int wib = __builtin_amdgcn_readfirstlane((int)(threadIdx.x >> 5));
  const int wavesPerBlock = blockDim.x >> 5;
  const int wave = blockIdx.x * wavesPerBlock + wib;
  const int nwaves = gridDim.x * wavesPerBlock;

  const long long* __restrict__ srcp = ei;
  const long long* __restrict__ dstp = ei + n_edges;

#pragma unroll 4
  for (int e = wave; e < n_edges; e += nwaves) {
    const int s = (int)srcp[e];
    const int d = (int)dstp[e];
    const float v = x[s * D + lane];
    atomicAdd(&F1[d * D + lane], v);
  }
}

// ---------------------------------------------------------------------------
// Kernel 3: out = x @ W2^T + b2 + F1 @ W1^T  via V_WMMA_F32_16X16X4_F32.
// One wave = one 16-node x 32-feature tile.  8 waves (256 threads) per block.
// A frag (16x4 f32): lane m=lane&15 is row, lane>>4 picks K pair {k0,k0+1}
//   vs {k0+2,k0+3}.  B = W^T, so B(k,n) = W[n][k] -> per-lane float2 from the
//   LDS copy of W at row (ntile*16 + (lane&15)), contiguous in k.
// C/D: VGPR r holds row r (lanes 0-15) / r+8 (lanes 16-31), col = lane&15.
// ---------------------------------------------------------------------------
__global__ __launch_bounds__(256) void fused_gemm(
    const float* __restrict__ x,
    const float* __restrict__ F1,
    const float* __restrict__ W1,
    const float* __restrict__ W2,
    const float* __restrict__ b2,
    float* __restrict__ out,
    int n_tiles) {
  __shared__ float w2s[D * D];
  __shared__ float w1s[D * D];
  __shared__ float b2s[D];
  __shared__ float xt[8][16 * D];
  __shared__ float ft[8][16 * D];

  const int tid = threadIdx.x;

  // Stage weights + bias into LDS (coalesced float4)
  ((float4*)w2s)[tid] = ((const float4*)W2)[tid];   // 256 * float4 = 1024 f
  ((float4*)w1s)[tid] = ((const float4*)W1)[tid];
  if (tid < 8) ((float4*)b2s)[tid] = ((const float4*)b2)[tid];
  __syncthreads();

  const int lane = tid & 31;
  const int w = tid >> 5;
  const int tile = blockIdx.x * 8 + w;
  const bool valid = tile < n_tiles;
  const int tb = valid ? tile * 16 : 0;   // dummy tile keeps EXEC/barriers uniform

  // Stage this wave's 16x32 x and F1 tiles (512 floats each = 128 float4)
  const float4* __restrict__ xg = (const float4*)(x + (size_t)tb * D);
  const float4* __restrict__ fg = (const float4*)(F1 + (size_t)tb * D);
  float4* xl = (float4*)xt[w];
  float4* fl = (float4*)ft[w];
#pragma unroll
  for (int i = 0; i < 4; ++i) {
    xl[i * 32 + lane] = xg[i * 32 + lane];
    fl[i * 32 + lane] = fg[i * 32 + lane];
  }
  __syncthreads();

  const int m = lane & 15;
  const int kh = (lane >> 4) << 1;   // 0 or 2: which K pair this half-wave owns

  v8f acc0 = {};   // N-tile 0 (features 0..15)
  v8f acc1 = {};   // N-tile 1 (features 16..31)

  const float* __restrict__ xrow = &xt[w][m * D];
  const float* __restrict__ frow = &ft[w][m * D];
  const float* __restrict__ w2r0 = &w2s[m * D];
  const float* __restrict__ w2r1 = &w2s[(16 + m) * D];
  const float* __restrict__ w1r0 = &w1s[m * D];
  const float* __restrict__ w1r1 = &w1s[(16 + m) * D];

#pragma unroll
  for (int kk = 0; kk < 8; ++kk) {
    const int k0 = kk * 4 + kh;
    v2f a  = *(const v2f*)(xrow + k0);
    v2f b0 = *(const v2f*)(w2r0 + k0);
    v2f b1 = *(const v2f*)(w2r1 + k0);
    acc0 = __builtin_amdgcn_wmma_f32_16x16x4_f32(false, a, false, b0, (short)0,
                                                 acc0, false, false);
    acc1 = __builtin_amdgcn_wmma_f32_16x16x4_f32(false, a, false, b1, (short)0,
                                                 acc1, false, false);
  }
#pragma unroll
  for (int kk = 0; kk < 8; ++kk) {
    const int k0 = kk * 4 + kh;
    v2f a  = *(const v2f*)(frow + k0);
    v2f b0 = *(const v2f*)(w1r0 + k0);
    v2f b1 = *(const v2f*)(w1r1 + k0);
    acc0 = __builtin_amdgcn_wmma_f32_16x16x4_f32(false, a, false, b0, (short)0,
                                                 acc0, false, false);
    acc1 = __builtin_amdgcn_wmma_f32_16x16x4_f32(false, a, false, b1, (short)0,
                                                 acc1, false, false);
  }

  const float bias0 = b2s[m];
  const float bias1 = b2s[16 + m];

  if (valid) {
    const int rbase = tb + ((lane >> 4) << 3);   // +8 rows for upper half-wave
#pragma unroll
    for (int r = 0; r < 8; ++r) {
      out[(size_t)(rbase + r) * D + m]      = acc0[r] + bias0;
      out[(size_t)(rbase + r) * D + 16 + m] = acc1[r] + bias1;
    }
  }
}

// ---------------------------------------------------------------------------
extern "C" void kernel_launch(void* const* d_in, const int* in_sizes, int n_in,
                              void* d_out, int out_size, void* d_ws, size_t ws_size,
                              hipStream_t stream) {
  const float* x       = (const float*)d_in[0];
  const float* W1      = (const float*)d_in[1];
  const float* W2      = (const float*)d_in[2];
  const float* b2      = (const float*)d_in[3];
  const long long* ei  = (const long long*)d_in[4];
  float* out = (float*)d_out;
  float* F1  = (float*)d_ws;

  const int n_nodes = in_sizes[0] / D;       // 100000
  const int n_edges = in_sizes[4] / 2;       // 1,600,000
  const int n4 = (n_nodes * D) / 4;

  // 1) F1 = 0
  zero_f32<<<(n4 + 255) / 256, 256, 0, stream>>>((float4*)F1, n4);

  // 2) F1[dst] += x[src]   (8192 waves, ~195 edges each)
  edge_scatter<<<1024, 256, 0, stream>>>(x, ei, F1, n_edges);

  // 3) out = x@W2^T + b2 + F1@W1^T   (6250 tiles, 8 per block)
  const int n_tiles = n_nodes / 16;
  const int blocks = (n_tiles + 7) / 8;
  fused_gemm<<<blocks, 256, 0, stream>>>(x, F1, W1, W2, b2, out, n_tiles);
}